// TangentSpaceLayer_64141041598486
// MI455X (gfx1250) — compile-verified
//
#include <hip/hip_runtime.h>
#include <hip/hip_bf16.h>
#include <math.h>

#define TPB     256
#define NSWEEP  4
#define EPSR    1e-8f

typedef int v4i_t __attribute__((ext_vector_type(4)));
typedef __attribute__((address_space(1))) v4i_t* gptr_v4i;
typedef __attribute__((address_space(3))) v4i_t* lptr_v4i;

// ---------------------------------------------------------------------------
// CDNA5 async global->LDS copy (per-lane b128), with portable fallbacks.
// ---------------------------------------------------------------------------
__device__ __forceinline__ void async_ld_f4(const float* gsrc, float* ldst) {
#if defined(__gfx1250__) && __has_builtin(__builtin_amdgcn_global_load_async_to_lds_b128)
    __builtin_amdgcn_global_load_async_to_lds_b128(
        (gptr_v4i)(void*)(const_cast<float*>(gsrc)),
        (lptr_v4i)(void*)ldst,
        /*imm offset*/ 0, /*cpol*/ 0);
#else
    *(float4*)ldst = *(const float4*)gsrc;
#endif
}

__device__ __forceinline__ void wait_async0() {
#if defined(__gfx1250__)
#if __has_builtin(__builtin_amdgcn_s_wait_asynccnt)
    __builtin_amdgcn_s_wait_asynccnt(0);
#else
    asm volatile("s_wait_asynccnt 0" ::: "memory");
#endif
#endif
}

// ---------------------------------------------------------------------------
// Fast hardware transcendentals (v_rcp_f32 / v_rsq_f32 / v_sqrt_f32, ~1 ulp).
// ---------------------------------------------------------------------------
__device__ __forceinline__ float frcp(float x) {
#if __has_builtin(__builtin_amdgcn_rcpf)
    return __builtin_amdgcn_rcpf(x);
#else
    return 1.0f / x;
#endif
}
__device__ __forceinline__ float frsq(float x) {
#if __has_builtin(__builtin_amdgcn_rsqf)
    return __builtin_amdgcn_rsqf(x);
#else
    return rsqrtf(x);
#endif
}
__device__ __forceinline__ float fsqrt(float x) {
#if __has_builtin(__builtin_amdgcn_sqrtf)
    return __builtin_amdgcn_sqrtf(x);
#else
    return sqrtf(x);
#endif
}

// ---------------------------------------------------------------------------
// One Jacobi rotation on the (P,Q) plane of a 4x4 symmetric matrix held in
// registers; R1,R2 are the other two indices. Branchless and overflow-safe:
// apq==0  ->  theta=+/-inf -> rcp(inf)=0 -> t=0 -> identity rotation.
// ---------------------------------------------------------------------------
template<int P, int Q, int R1, int R2>
__device__ __forceinline__ void jrot(float A[4][4], float V[4][4]) {
    const float apq  = A[P][Q];
    const float den  = 2.0f * apq;
    const float theta = (A[Q][Q] - A[P][P]) * frcp(den + copysignf(1e-30f, den));
    const float t = copysignf(1.0f, theta) *
                    frcp(fabsf(theta) + fsqrt(fmaf(theta, theta, 1.0f)));
    const float c = frsq(fmaf(t, t, 1.0f));
    const float s = t * c;

    // Diagonal + annihilated off-diagonal
    A[P][P] = fmaf(-t, apq, A[P][P]);
    A[Q][Q] = fmaf( t, apq, A[Q][Q]);
    A[P][Q] = 0.0f;
    A[Q][P] = 0.0f;

    // Remaining off-block entries (keep full symmetric storage coherent)
    {
        const float akp = A[R1][P], akq = A[R1][Q];
        const float np = fmaf(c, akp, -s * akq);
        const float nq = fmaf(s, akp,  c * akq);
        A[R1][P] = np; A[P][R1] = np;
        A[R1][Q] = nq; A[Q][R1] = nq;
    }
    {
        const float akp = A[R2][P], akq = A[R2][Q];
        const float np = fmaf(c, akp, -s * akq);
        const float nq = fmaf(s, akp,  c * akq);
        A[R2][P] = np; A[P][R2] = np;
        A[R2][Q] = nq; A[Q][R2] = nq;
    }

    // Accumulate eigenvectors: V <- V * J
#pragma unroll
    for (int k = 0; k < 4; ++k) {
        const float vkp = V[k][P], vkq = V[k][Q];
        V[k][P] = fmaf(c, vkp, -s * vkq);
        V[k][Q] = fmaf(s, vkp,  c * vkq);
    }
}

// ---------------------------------------------------------------------------
// Kernel: one 4x4 SPD matrix per lane, 256 matrices per block.
//   Phase 1: async-stage 256*64B of input into LDS (coalesced b128 lanes)
//   Phase 2: per-lane symmetrize + Jacobi eigensolve + log-reconstruct
//   Phase 3: stage 10 outputs/matrix in LDS, write back fully coalesced
// ---------------------------------------------------------------------------
__global__ __launch_bounds__(TPB) void tangent_space_logm_kernel(
        const float* __restrict__ x, float* __restrict__ out, int nmat) {
    // Input: pad each matrix to 20 floats (80 B) -> stride 20 words, few conflicts
    __shared__ float ldsIn[TPB * 20];     // 20 KB
    __shared__ float ldsOut[TPB * 10];    // 10 KB

    const int tid = threadIdx.x;
    const int m0  = blockIdx.x * TPB;               // first matrix of this block
    const int valid = min(TPB, nmat - m0);          // matrices handled here
    const int total4 = nmat * 4;                    // total float4 count of input

    // ---- Phase 1: async global -> LDS, 4 x b128 per lane, coalesced ----
#pragma unroll
    for (int k = 0; k < 4; ++k) {
        const int flat4 = tid + k * TPB;            // 0..1023 within chunk
        const int g4    = m0 * 4 + flat4;           // global float4 index
        if (g4 < total4) {
            const int m    = flat4 >> 2;            // local matrix
            const int part = flat4 & 3;             // which row (float4)
            async_ld_f4(x + (size_t)g4 * 4, &ldsIn[m * 20 + part * 4]);
        }
    }
    wait_async0();
    __syncthreads();

    // ---- Phase 2: per-lane eigensolve ----
    if (tid < valid) {
        const float* xm = &ldsIn[tid * 20];

        float A[4][4];
        float V[4][4];
#pragma unroll
        for (int i = 0; i < 4; ++i) {
#pragma unroll
            for (int j = 0; j < 4; ++j) {
                float v = 0.5f * (xm[i * 4 + j] + xm[j * 4 + i]);
                if (i == j) v += EPSR;
                A[i][j] = v;
                V[i][j] = (i == j) ? 1.0f : 0.0f;
            }
        }

        // Fully unrolled so the first sweep folds V==identity to constants.
#pragma unroll
        for (int sw = 0; sw < NSWEEP; ++sw) {
            jrot<0, 1, 2, 3>(A, V);
            jrot<0, 2, 1, 3>(A, V);
            jrot<0, 3, 1, 2>(A, V);
            jrot<1, 2, 0, 3>(A, V);
            jrot<1, 3, 0, 2>(A, V);
            jrot<2, 3, 0, 1>(A, V);
        }

        // log of eigenvalues, W[i][k] = V[i][k] * log(w_k)
        float W[4][4];
#pragma unroll
        for (int k = 0; k < 4; ++k) {
            const float lw = __logf(fmaxf(A[k][k], EPSR));
#pragma unroll
            for (int i = 0; i < 4; ++i) W[i][k] = V[i][k] * lw;
        }

        // logm[i][j] = sum_k W[i][k] * V[j][k]; emit upper triangle row-major
        float* o = &ldsOut[tid * 10];
        int n = 0;
#pragma unroll
        for (int i = 0; i < 4; ++i) {
#pragma unroll
            for (int j = i; j < 4; ++j) {
                float acc = W[i][0] * V[j][0];
                acc = fmaf(W[i][1], V[j][1], acc);
                acc = fmaf(W[i][2], V[j][2], acc);
                acc = fmaf(W[i][3], V[j][3], acc);
                o[n++] = acc;
            }
        }
    }
    __syncthreads();

    // ---- Phase 3: coalesced writeback (flat LDS layout == flat global) ----
    const int validf = valid * 10;
    const size_t obase = (size_t)m0 * 10;
#pragma unroll
    for (int r = 0; r < 10; ++r) {
        const int li = r * TPB + tid;
        if (li < validf) out[obase + li] = ldsOut[li];
    }
}

// ---------------------------------------------------------------------------
extern "C" void kernel_launch(void* const* d_in, const int* in_sizes, int n_in,
                              void* d_out, int out_size, void* d_ws, size_t ws_size,
                              hipStream_t stream) {
    (void)n_in; (void)out_size; (void)d_ws; (void)ws_size;
    const float* x = (const float*)d_in[0];
    float* out = (float*)d_out;
    const int nmat = in_sizes[0] / 16;
    if (nmat <= 0) return;
    const int blocks = (nmat + TPB - 1) / TPB;
    tangent_space_logm_kernel<<<blocks, TPB, 0, stream>>>(x, out, nmat);
}